// PoreGraphGNN_6339371729326
// MI455X (gfx1250) — compile-verified
//
#include <hip/hip_runtime.h>
#include <hip/hip_bf16.h>
#include <math.h>

typedef __attribute__((ext_vector_type(2))) float v2f;
typedef __attribute__((ext_vector_type(8))) float v8f;

#define NN 100000
#define FF 20
#define HH 128
#define EE 1600000
#define GG 64
#define LL 4

// ---------------------------------------------------------------------------
// x = tanh(voro_x @ W_in)   [N,20] @ [20,128]
// ---------------------------------------------------------------------------
#define EMB_NODES 16
__global__ void __launch_bounds__(256) embed_kernel(const float* __restrict__ voro,
                                                    const float* __restrict__ Win,
                                                    float* __restrict__ x, int nrows) {
    __shared__ float sW[FF * HH];  // 10 KB
    for (int i = threadIdx.x; i < FF * HH; i += 256) sW[i] = Win[i];
    __syncthreads();
    int base = blockIdx.x * EMB_NODES;
    for (int t = threadIdx.x; t < EMB_NODES * HH; t += 256) {
        int n = base + (t >> 7);
        int c = t & 127;
        if (n < nrows) {
            const float* vr = voro + (size_t)n * FF;
            float acc = 0.f;
#pragma unroll
            for (int f = 0; f < FF; ++f) acc += vr[f] * sW[f * HH + c];
            x[(size_t)n * HH + c] = tanhf(acc);
        }
    }
}

// ---------------------------------------------------------------------------
// zero fill
// ---------------------------------------------------------------------------
__global__ void zero_kernel(float* __restrict__ p, long long n) {
    long long i = (long long)blockIdx.x * blockDim.x + threadIdx.x;
    long long stride = (long long)gridDim.x * blockDim.x;
    for (; i < n; i += stride) p[i] = 0.f;
}

// ---------------------------------------------------------------------------
// m = x @ W   [N,128] @ [128,128]  via V_WMMA_F32_16X16X4_F32
// block: 256 thr = 8 waves; block covers 16 rows, wave w covers cols [16w,16w+16)
// ---------------------------------------------------------------------------
__global__ void __launch_bounds__(256) mpnn_kernel(const float* __restrict__ x,
                                                   const float* __restrict__ W,
                                                   float* __restrict__ m, int nrows) {
    __shared__ float sW[HH * HH];   // 64 KB
    __shared__ float sA[16 * HH];   // 8 KB
    for (int i = threadIdx.x; i < HH * HH; i += 256) sW[i] = W[i];
    int row0 = blockIdx.x * 16;
    for (int i = threadIdx.x; i < 16 * HH; i += 256) {
        int r = row0 + (i >> 7);
        sA[i] = (r < nrows) ? x[(size_t)r * HH + (i & 127)] : 0.f;
    }
    __syncthreads();

    const int wave = threadIdx.x >> 5;
    const int lane = threadIdx.x & 31;
    const int colbase = wave * 16;
    const int lm = lane & 15;
    const int rsel = lane >> 4;        // 0 for lanes 0-15, 1 for lanes 16-31
    const int ksel = rsel * 2;

    v8f c = {};
#pragma unroll
    for (int k = 0; k < HH; k += 4) {
        v2f a, b;
        a.x = sA[lm * HH + k + ksel];
        a.y = sA[lm * HH + k + ksel + 1];
        b.x = sW[(k + ksel) * HH + colbase + lm];
        b.y = sW[(k + ksel + 1) * HH + colbase + lm];
        c = __builtin_amdgcn_wmma_f32_16x16x4_f32(false, a, false, b, (short)0, c, false, false);
    }
#pragma unroll
    for (int vg = 0; vg < 8; ++vg) {
        int r = row0 + vg + 8 * rsel;
        if (r < nrows) m[(size_t)r * HH + colbase + lm] = c[vg];
    }
}

// ---------------------------------------------------------------------------
// agg[dst] += m[src]  over E edges; 1 wave per edge, float4 per lane
// ---------------------------------------------------------------------------
__global__ void __launch_bounds__(256) scatter_kernel(const float* __restrict__ m,
                                                      const int* __restrict__ ei,
                                                      float* __restrict__ agg, int ne) {
    long long t = (long long)blockIdx.x * blockDim.x + threadIdx.x;
    int e = (int)(t >> 5);
    if (e >= ne) return;
    int q = (int)(t & 31) * 4;
    int src = ei[e];
    int dst = ei[ne + e];
    const float4 v = *(const float4*)(m + (size_t)src * HH + q);
    float* d = agg + (size_t)dst * HH + q;
    atomicAdd(d + 0, v.x);
    atomicAdd(d + 1, v.y);
    atomicAdd(d + 2, v.z);
    atomicAdd(d + 3, v.w);
}

// ---------------------------------------------------------------------------
// Fused GRU cell: h_out = GRU(agg, h)
// gi = agg @ w_ih^T + b_ih ; gh = h @ w_hh^T + b_hh ; gate math ; blend
// block covers 16 rows; wave w covers output cols [16w,16w+16); 6 WMMA accums.
// w^T makes K contiguous per output row -> adjacent-pair (b64) weight loads.
// ---------------------------------------------------------------------------
__global__ void __launch_bounds__(256) gru_kernel(const float* __restrict__ agg,
                                                  const float* __restrict__ h,
                                                  const float* __restrict__ wih,
                                                  const float* __restrict__ whh,
                                                  const float* __restrict__ bih,
                                                  const float* __restrict__ bhh,
                                                  float* __restrict__ hout, int nrows) {
    __shared__ float sAgg[16 * HH];  // 8 KB
    __shared__ float sH[16 * HH];    // 8 KB
    int row0 = blockIdx.x * 16;
    for (int i = threadIdx.x; i < 16 * HH; i += 256) {
        int r = row0 + (i >> 7);
        bool ok = (r < nrows);
        sAgg[i] = ok ? agg[(size_t)r * HH + (i & 127)] : 0.f;
        sH[i]   = ok ? h[(size_t)r * HH + (i & 127)]   : 0.f;
    }
    __syncthreads();

    const int wave = threadIdx.x >> 5;
    const int lane = threadIdx.x & 31;
    const int colbase = wave * 16;
    const int lm = lane & 15;
    const int rsel = lane >> 4;
    const int ksel = rsel * 2;

    // per-lane weight row pointers (output row = gate_off + colbase + lm)
    const float* pir = wih + (size_t)(0 * HH + colbase + lm) * HH;
    const float* piz = wih + (size_t)(1 * HH + colbase + lm) * HH;
    const float* pin = wih + (size_t)(2 * HH + colbase + lm) * HH;
    const float* phr = whh + (size_t)(0 * HH + colbase + lm) * HH;
    const float* phz = whh + (size_t)(1 * HH + colbase + lm) * HH;
    const float* phn = whh + (size_t)(2 * HH + colbase + lm) * HH;

    v8f cir = {}, ciz = {}, cin = {}, chr_ = {}, chz = {}, chn = {};
#pragma unroll
    for (int k = 0; k < HH; k += 4) {
        v2f aA, aH, b;
        aA.x = sAgg[lm * HH + k + ksel]; aA.y = sAgg[lm * HH + k + ksel + 1];
        aH.x = sH[lm * HH + k + ksel];   aH.y = sH[lm * HH + k + ksel + 1];

        b.x = pir[k + ksel]; b.y = pir[k + ksel + 1];
        cir = __builtin_amdgcn_wmma_f32_16x16x4_f32(false, aA, false, b, (short)0, cir, false, false);
        b.x = piz[k + ksel]; b.y = piz[k + ksel + 1];
        ciz = __builtin_amdgcn_wmma_f32_16x16x4_f32(false, aA, false, b, (short)0, ciz, false, false);
        b.x = pin[k + ksel]; b.y = pin[k + ksel + 1];
        cin = __builtin_amdgcn_wmma_f32_16x16x4_f32(false, aA, false, b, (short)0, cin, false, false);
        b.x = phr[k + ksel]; b.y = phr[k + ksel + 1];
        chr_ = __builtin_amdgcn_wmma_f32_16x16x4_f32(false, aH, false, b, (short)0, chr_, false, false);
        b.x = phz[k + ksel]; b.y = phz[k + ksel + 1];
        chz = __builtin_amdgcn_wmma_f32_16x16x4_f32(false, aH, false, b, (short)0, chz, false, false);
        b.x = phn[k + ksel]; b.y = phn[k + ksel + 1];
        chn = __builtin_amdgcn_wmma_f32_16x16x4_f32(false, aH, false, b, (short)0, chn, false, false);
    }

    const int col = colbase + lm;
    const float bir = bih[0 * HH + col], biz = bih[1 * HH + col], bin_ = bih[2 * HH + col];
    const float bhr = bhh[0 * HH + col], bhz = bhh[1 * HH + col], bhn = bhh[2 * HH + col];

#pragma unroll
    for (int vg = 0; vg < 8; ++vg) {
        int lr = vg + 8 * rsel;
        int r = row0 + lr;
        if (r < nrows) {
            float ir = cir[vg] + bir, hr = chr_[vg] + bhr;
            float iz = ciz[vg] + biz, hz = chz[vg] + bhz;
            float in_ = cin[vg] + bin_, hn = chn[vg] + bhn;
            float rg = 1.f / (1.f + expf(-(ir + hr)));
            float zg = 1.f / (1.f + expf(-(iz + hz)));
            float ng = tanhf(in_ + rg * hn);
            float hold = sH[lr * HH + col];
            hout[(size_t)r * HH + col] = (1.f - zg) * ng + zg * hold;
        }
    }
}

// ---------------------------------------------------------------------------
// segment sums over sorted batch: register-accumulate runs, flush per g-change
// ---------------------------------------------------------------------------
#define POOL_NODES 64
__global__ void __launch_bounds__(256) pool_kernel(const float* __restrict__ x,
                                                   const int* __restrict__ batch,
                                                   float* __restrict__ sums,
                                                   float* __restrict__ cnts, int nrows) {
    int base = blockIdx.x * POOL_NODES;
    int end = base + POOL_NODES; if (end > nrows) end = nrows;
    int c = threadIdx.x & 127;
    int p = threadIdx.x >> 7;  // 0..1
    float acc = 0.f; int curg = -1; int cnt = 0;
    for (int n = base + p; n < end; n += 2) {
        int g = batch[n];
        if (g != curg) {
            if (curg >= 0) {
                atomicAdd(&sums[curg * HH + c], acc);
                if (c == 0) atomicAdd(&cnts[curg], (float)cnt);
            }
            curg = g; acc = 0.f; cnt = 0;
        }
        acc += x[(size_t)n * HH + c];
        cnt++;
    }
    if (curg >= 0) {
        atomicAdd(&sums[curg * HH + c], acc);
        if (c == 0) atomicAdd(&cnts[curg], (float)cnt);
    }
}

// ---------------------------------------------------------------------------
// out[g] = relu(sums[g]/max(cnt,1)) . W_pred + b_pred
// ---------------------------------------------------------------------------
__global__ void __launch_bounds__(128) pred_kernel(const float* __restrict__ sums,
                                                   const float* __restrict__ cnts,
                                                   const float* __restrict__ Wp,
                                                   const float* __restrict__ bp,
                                                   float* __restrict__ out) {
    __shared__ float red[HH];
    int g = blockIdx.x;
    int c = threadIdx.x;
    float denom = fmaxf(cnts[g], 1.f);
    float v = fmaxf(sums[g * HH + c] / denom, 0.f);
    red[c] = v * Wp[c];
    __syncthreads();
    for (int s = 64; s > 0; s >>= 1) {
        if (c < s) red[c] += red[c + s];
        __syncthreads();
    }
    if (c == 0) out[g] = red[0] + bp[0];
}

// ---------------------------------------------------------------------------
extern "C" void kernel_launch(void* const* d_in, const int* in_sizes, int n_in,
                              void* d_out, int out_size, void* d_ws, size_t ws_size,
                              hipStream_t stream) {
    (void)in_sizes; (void)n_in; (void)out_size; (void)ws_size;

    const float* voro  = (const float*)d_in[0];
    const int*   edge  = (const int*)d_in[1];
    const int*   batch = (const int*)d_in[2];
    const float* Win   = (const float*)d_in[3];
    const float* mpnnW = (const float*)d_in[4];
    const float* wih   = (const float*)d_in[5];
    const float* whh   = (const float*)d_in[6];
    const float* bih   = (const float*)d_in[7];
    const float* bhh   = (const float*)d_in[8];
    const float* Wp    = (const float*)d_in[9];
    const float* bp    = (const float*)d_in[10];
    float* out = (float*)d_out;

    char* ws = (char*)d_ws;
    size_t bufBytes = (size_t)NN * HH * sizeof(float);   // 51.2 MB
    float* bufA   = (float*)(ws);
    float* bufB   = (float*)(ws + bufBytes);
    float* bufAgg = (float*)(ws + 2 * bufBytes);
    float* sums   = (float*)(ws + 3 * bufBytes);         // G*H
    float* cnts   = sums + GG * HH;                      // G (contiguous after sums)

    const int rowBlocks = (NN + 15) / 16;                // 6250

    embed_kernel<<<(NN + EMB_NODES - 1) / EMB_NODES, 256, 0, stream>>>(voro, Win, bufA, NN);

    float* xcur = bufA;
    float* xnext = bufB;
    for (int l = 0; l < LL; ++l) {
        mpnn_kernel<<<rowBlocks, 256, 0, stream>>>(xcur, mpnnW + (size_t)l * HH * HH, xnext, NN);
        zero_kernel<<<2048, 256, 0, stream>>>(bufAgg, (long long)NN * HH);
        scatter_kernel<<<(int)(((long long)EE * 32 + 255) / 256), 256, 0, stream>>>(xnext, edge, bufAgg, EE);
        // m (in xnext) is dead after scatter: write h_out over it, then swap
        gru_kernel<<<rowBlocks, 256, 0, stream>>>(bufAgg, xcur, wih, whh, bih, bhh, xnext, NN);
        float* tmp = xcur; xcur = xnext; xnext = tmp;
    }

    zero_kernel<<<40, 256, 0, stream>>>(sums, (long long)(GG * HH + GG));
    pool_kernel<<<(NN + POOL_NODES - 1) / POOL_NODES, 256, 0, stream>>>(xcur, batch, sums, cnts, NN);
    pred_kernel<<<GG, 128, 0, stream>>>(sums, cnts, Wp, bp, out);
}